// GraphAT_28063316313036
// MI455X (gfx1250) — compile-verified
//
#include <hip/hip_runtime.h>
#include <math.h>

typedef __attribute__((ext_vector_type(2))) float v2f;
typedef __attribute__((ext_vector_type(8))) float v8f;

#define GN 4096
#define IN_F 256
#define NWORDS (GN / 32)
#define ALPHA 0.2f
#define NEGV (-9e15f)

__device__ __forceinline__ float lrelu(float x) { return x > 0.f ? x : ALPHA * x; }
__device__ __forceinline__ float eluf(float x) { return x > 0.f ? x : expm1f(x); }

// ---------------------------------------------------------------------------
// Pack adj (int32 0/1, 64MB) into a bitmask (2MB) that stays L2-resident.
// ---------------------------------------------------------------------------
__global__ void pack_adj(const int* __restrict__ adj, unsigned* __restrict__ bits) {
    int w = blockIdx.x * blockDim.x + threadIdx.x;
    if (w >= GN * NWORDS) return;
    int row = w / NWORDS, word = w % NWORDS;
    const int* p = adj + (size_t)row * GN + (size_t)word * 32;
    unsigned v = 0;
#pragma unroll
    for (int b = 0; b < 32; ++b) v |= (p[b] > 0) ? (1u << b) : 0u;
    bits[w] = v;
}

// ---------------------------------------------------------------------------
// C[M,ncols] = X[M,256] @ W[256,ncols] via v_wmma_f32_16x16x4_f32.
// One wave per 16x16 output tile; blockIdx.z = batch (head).
// ---------------------------------------------------------------------------
__global__ void gemm_wmma(const float* __restrict__ X, const float* __restrict__ W,
                          float* __restrict__ Out, int ldw, int ldo,
                          long long wBatchStride, long long oBatchStride) {
    int lane = threadIdx.x & 31;
    int wave = threadIdx.x >> 5;
    int row0 = blockIdx.x * 16;
    int col0 = (blockIdx.y * 4 + wave) * 16;
    const float* Wb = W + (size_t)blockIdx.z * wBatchStride;
    float* Ob = Out + (size_t)blockIdx.z * oBatchStride;
    int m = lane & 15;          // row (A) / col (B) within tile
    int kb = (lane >> 4) * 2;   // K sub-offset for this half-wave
    const float* xrow = X + (size_t)(row0 + m) * IN_F;
    v8f c;
#pragma unroll
    for (int v = 0; v < 8; ++v) c[v] = 0.f;
    for (int k = 0; k < IN_F; k += 4) {
        v2f a = *(const v2f*)(xrow + k + kb);
        v2f b;
        b.x = Wb[(size_t)(k + kb) * ldw + col0 + m];
        b.y = Wb[(size_t)(k + kb + 1) * ldw + col0 + m];
        c = __builtin_amdgcn_wmma_f32_16x16x4_f32(false, a, false, b, (short)0, c,
                                                  false, false);
    }
#pragma unroll
    for (int v = 0; v < 8; ++v) {
        int r = v + (lane >> 4) * 8;
        Ob[(size_t)(row0 + r) * ldo + col0 + m] = c[v];
    }
}

// ---------------------------------------------------------------------------
// s1[i] = Wh[i,:] @ a[:F],  s2[i] = Wh[i,:] @ a[F:]
// ---------------------------------------------------------------------------
__global__ void svec(const float* __restrict__ Wh, const float* __restrict__ a,
                     float* __restrict__ s1, float* __restrict__ s2, int F,
                     long long whStride, long long aStride) {
    int i = blockIdx.x * blockDim.x + threadIdx.x;
    int bat = blockIdx.y;
    if (i >= GN) return;
    const float* wh = Wh + (size_t)bat * whStride + (size_t)i * F;
    const float* av = a + (size_t)bat * aStride;
    float acc1 = 0.f, acc2 = 0.f;
    for (int f = 0; f < F; ++f) {
        float w = wh[f];
        acc1 += w * av[f];
        acc2 += w * av[F + f];
    }
    s1[(size_t)bat * GN + i] = acc1;
    s2[(size_t)bat * GN + i] = acc2;
}

// ---------------------------------------------------------------------------
// Per-row softmax stats: m_i = max_j val_ij, isum_i = 1/sum_j exp(val_ij - m_i)
// val_ij = mask ? leakyrelu(s1_i + s2_j) : NEG.  One wave per row.
// ---------------------------------------------------------------------------
__global__ void rowstats(const unsigned* __restrict__ bits, const float* __restrict__ s1,
                         const float* __restrict__ s2, float* __restrict__ mOut,
                         float* __restrict__ isumOut) {
    int lane = threadIdx.x & 31;
    int wave = threadIdx.x >> 5;
    int i = blockIdx.x * (blockDim.x >> 5) + wave;
    int bat = blockIdx.y;
    const float* s2b = s2 + (size_t)bat * GN;
    float s1i = s1[(size_t)bat * GN + i];
    const unsigned* brow = bits + (size_t)i * NWORDS;
    float mx = NEGV;
    for (int t = 0; t < NWORDS; ++t) {
        unsigned w = brow[t];
        float v = ((w >> lane) & 1u) ? lrelu(s1i + s2b[t * 32 + lane]) : NEGV;
        mx = fmaxf(mx, v);
    }
    for (int o = 16; o; o >>= 1) mx = fmaxf(mx, __shfl_xor(mx, o, 32));
    float sum = 0.f;
    for (int t = 0; t < NWORDS; ++t) {
        unsigned w = brow[t];
        float v = ((w >> lane) & 1u) ? lrelu(s1i + s2b[t * 32 + lane]) : NEGV;
        sum += __expf(v - mx);
    }
    for (int o = 16; o; o >>= 1) sum += __shfl_xor(sum, o, 32);
    if (lane == 0) {
        mOut[(size_t)bat * GN + i] = mx;
        isumOut[(size_t)bat * GN + i] = 1.f / sum;
    }
}

// ---------------------------------------------------------------------------
// weights[j] = sum_i att[i,j]  (column sums of layer-2 attention), wave per j.
// ---------------------------------------------------------------------------
__global__ void colsum(const unsigned* __restrict__ bits, const float* __restrict__ s1,
                       const float* __restrict__ s2, const float* __restrict__ mrow,
                       const float* __restrict__ isum, float* __restrict__ weights) {
    int lane = threadIdx.x & 31;
    int wave = threadIdx.x >> 5;
    int j = blockIdx.x * (blockDim.x >> 5) + wave;
    float s2j = s2[j];
    int word = j >> 5, bit = j & 31;
    float acc = 0.f;
    for (int i = lane; i < GN; i += 32) {
        unsigned w = bits[(size_t)i * NWORDS + word];
        float v = ((w >> bit) & 1u) ? lrelu(s1[i] + s2j) : NEGV;
        acc += __expf(v - mrow[i]) * isum[i];
    }
    for (int o = 16; o; o >>= 1) acc += __shfl_xor(acc, o, 32);
    if (lane == 0) weights[j] = acc;
}

// ---------------------------------------------------------------------------
// Fused flash-style softmax @ Wh: block owns 32 rows (two 16-row WMMA groups
// sharing every B load -> halves L2 Wh traffic), 4 waves split the j range,
// p computed on the fly into WMMA A-operands, f32 accumulate, deterministic
// two-stage LDS reduce, scale by 1/sum, ELU (+ optional per-row weight scale).
// ---------------------------------------------------------------------------
template <int NCOLS>
__global__ void att_matmul(const unsigned* __restrict__ bits, const float* __restrict__ s1,
                           const float* __restrict__ s2, const float* __restrict__ mrow,
                           const float* __restrict__ isum, const float* __restrict__ Wh,
                           float* __restrict__ out, int ldo, int obatColStep,
                           const float* __restrict__ rowscale) {
    constexpr int NT = NCOLS / 16;
    __shared__ float red[2][32][NCOLS];
    int lane = threadIdx.x & 31;
    int wave = threadIdx.x >> 5;
    int i0 = blockIdx.x * 32;
    int bat = blockIdx.z;
    const float* s2b = s2 + (size_t)bat * GN;
    const float* whb = Wh + (size_t)bat * GN * NCOLS;
    int m = lane & 15;
    int kb = (lane >> 4) * 2;
    int row0 = i0 + m, row1 = i0 + 16 + m;
    float s1i0 = s1[(size_t)bat * GN + row0];
    float mi0 = mrow[(size_t)bat * GN + row0];
    float s1i1 = s1[(size_t)bat * GN + row1];
    float mi1 = mrow[(size_t)bat * GN + row1];
    const unsigned* brow0 = bits + (size_t)row0 * NWORDS;
    const unsigned* brow1 = bits + (size_t)row1 * NWORDS;

    v8f c0[NT], c1[NT];
#pragma unroll
    for (int t = 0; t < NT; ++t)
#pragma unroll
        for (int v = 0; v < 8; ++v) { c0[t][v] = 0.f; c1[t][v] = 0.f; }

    int jbeg = wave * (GN / 4);
    int jend = jbeg + (GN / 4);
    for (int j0 = jbeg; j0 < jend; j0 += 4) {
        int widx = j0 >> 5;
        int bsh = (j0 & 31) + kb;
        unsigned wb0 = brow0[widx];
        unsigned wb1 = brow1[widx];
        int j = j0 + kb;
        v2f s2p = *(const v2f*)(s2b + j);
        float e00 = ((wb0 >> bsh) & 1u) ? lrelu(s1i0 + s2p.x) : NEGV;
        float e01 = ((wb0 >> (bsh + 1)) & 1u) ? lrelu(s1i0 + s2p.y) : NEGV;
        float e10 = ((wb1 >> bsh) & 1u) ? lrelu(s1i1 + s2p.x) : NEGV;
        float e11 = ((wb1 >> (bsh + 1)) & 1u) ? lrelu(s1i1 + s2p.y) : NEGV;
        v2f a0, a1;
        a0.x = __expf(e00 - mi0);  // unnormalized p_{row0, j..j+1}
        a0.y = __expf(e01 - mi0);
        a1.x = __expf(e10 - mi1);  // unnormalized p_{row1, j..j+1}
        a1.y = __expf(e11 - mi1);
        const float* bp0 = whb + (size_t)j * NCOLS;
        const float* bp1 = bp0 + NCOLS;
#pragma unroll
        for (int t = 0; t < NT; ++t) {
            v2f b;
            b.x = bp0[t * 16 + m];
            b.y = bp1[t * 16 + m];
            c0[t] = __builtin_amdgcn_wmma_f32_16x16x4_f32(false, a0, false, b, (short)0,
                                                          c0[t], false, false);
            c1[t] = __builtin_amdgcn_wmma_f32_16x16x4_f32(false, a1, false, b, (short)0,
                                                          c1[t], false, false);
        }
    }

    // Deterministic two-stage cross-wave reduction: odd waves store partials,
    // even waves accumulate theirs on top, then everyone combines the 2 banks.
    int bank = wave >> 1;
    if (wave & 1) {
#pragma unroll
        for (int t = 0; t < NT; ++t)
#pragma unroll
            for (int v = 0; v < 8; ++v) {
                int r = v + (lane >> 4) * 8;
                red[bank][r][t * 16 + m] = c0[t][v];
                red[bank][16 + r][t * 16 + m] = c1[t][v];
            }
    }
    __syncthreads();
    if (!(wave & 1)) {
#pragma unroll
        for (int t = 0; t < NT; ++t)
#pragma unroll
            for (int v = 0; v < 8; ++v) {
                int r = v + (lane >> 4) * 8;
                red[bank][r][t * 16 + m] += c0[t][v];
                red[bank][16 + r][t * 16 + m] += c1[t][v];
            }
    }
    __syncthreads();

    int col0 = bat * obatColStep;
    for (int idx = threadIdx.x; idx < 32 * NCOLS; idx += blockDim.x) {
        int r = idx / NCOLS, cc = idx % NCOLS;
        float sv = red[0][r][cc] + red[1][r][cc];
        int i = i0 + r;
        float val = eluf(sv * isum[(size_t)bat * GN + i]);
        if (rowscale) val *= rowscale[i];
        out[(size_t)i * ldo + col0 + cc] = val;
    }
}

// ---------------------------------------------------------------------------
extern "C" void kernel_launch(void* const* d_in, const int* in_sizes, int n_in,
                              void* d_out, int out_size, void* d_ws, size_t ws_size,
                              hipStream_t stream) {
    const float* x = (const float*)d_in[0];        // [4096,256]
    const int* adj = (const int*)d_in[1];          // [4096,4096]
    const float* Ws = (const float*)d_in[2];       // [4,256,64]
    const float* a_heads = (const float*)d_in[3];  // [4,128]
    const float* W_out = (const float*)d_in[4];    // [256,128]
    const float* a_out = (const float*)d_in[5];    // [256]
    float* out = (float*)d_out;                    // [1,4096,128]

    char* p = (char*)d_ws;
    unsigned* bitsw = (unsigned*)p; p += (size_t)GN * NWORDS * 4;
    float* Wh1 = (float*)p;  p += (size_t)4 * GN * 64 * 4;
    float* s1_1 = (float*)p; p += (size_t)4 * GN * 4;
    float* s2_1 = (float*)p; p += (size_t)4 * GN * 4;
    float* m1 = (float*)p;   p += (size_t)4 * GN * 4;
    float* is1 = (float*)p;  p += (size_t)4 * GN * 4;
    float* h = (float*)p;    p += (size_t)GN * 256 * 4;
    float* Wh2 = (float*)p;  p += (size_t)GN * 128 * 4;
    float* s1_2 = (float*)p; p += (size_t)GN * 4;
    float* s2_2 = (float*)p; p += (size_t)GN * 4;
    float* m2 = (float*)p;   p += (size_t)GN * 4;
    float* is2 = (float*)p;  p += (size_t)GN * 4;
    float* wcol = (float*)p; p += (size_t)GN * 4;

    // 0) adjacency -> bitmask (L2-resident for all later passes)
    pack_adj<<<(GN * NWORDS) / 256, 256, 0, stream>>>(adj, bitsw);

    // 1) Wh per head: x @ Ws[h]  -> Wh1[h][4096][64]
    gemm_wmma<<<dim3(GN / 16, 1, 4), 128, 0, stream>>>(
        x, Ws, Wh1, 64, 64, (long long)256 * 64, (long long)GN * 64);

    // 2) s1/s2 per head
    svec<<<dim3(GN / 256, 4), 256, 0, stream>>>(Wh1, a_heads, s1_1, s2_1, 64,
                                                (long long)GN * 64, 128);

    // 3) row max / inv-sum per head
    rowstats<<<dim3(GN / 4, 4), 128, 0, stream>>>(bitsw, s1_1, s2_1, m1, is1);

    // 4) fused softmax @ Wh1, ELU, concat into h[4096][256]
    att_matmul<64><<<dim3(GN / 32, 1, 4), 128, 0, stream>>>(
        bitsw, s1_1, s2_1, m1, is1, Wh1, h, 256, 64, nullptr);

    // 5) Wh2 = h @ W_out  -> [4096][128]
    gemm_wmma<<<dim3(GN / 16, 2, 1), 128, 0, stream>>>(h, W_out, Wh2, 128, 128, 0, 0);

    // 6) output-layer s1/s2
    svec<<<dim3(GN / 256, 1), 256, 0, stream>>>(Wh2, a_out, s1_2, s2_2, 128, 0, 0);

    // 7) output-layer row stats
    rowstats<<<dim3(GN / 4, 1), 128, 0, stream>>>(bitsw, s1_2, s2_2, m2, is2);

    // 8) column sums of layer-2 attention (the "weights" vector)
    colsum<<<dim3(GN / 4), 128, 0, stream>>>(bitsw, s1_2, s2_2, m2, is2, wcol);

    // 9) fused softmax @ Wh2, ELU, scale by weights -> d_out
    att_matmul<128><<<dim3(GN / 32, 1, 1), 128, 0, stream>>>(
        bitsw, s1_2, s2_2, m2, is2, Wh2, out, 128, 0, wcol);
}